// CrossEntropyLoss_63977832841845
// MI455X (gfx1250) — compile-verified
//
#include <hip/hip_runtime.h>
#include <hip/hip_bf16.h>

typedef float v2f __attribute__((ext_vector_type(2)));
typedef float v8f __attribute__((ext_vector_type(8)));

#define DCOLS   512   // feature dim (reference: D = 512)
#define TROWS   16    // rows per WMMA tile
#define CHUNK   64    // columns staged to LDS per step
#define LSTRIDE 68    // 64 + 4 pad dwords: %64==4 -> conflict-free b64 reads, rows 16B-aligned
#define WAVES   4
#define BLOCK   128

__device__ __forceinline__ float pick8(v8f v, int idx) {
  float r = v[0];
#pragma unroll
  for (int i = 1; i < 8; ++i) r = (idx == i) ? v[i] : r;
  return r;
}

// One wave processes a 16-row tile. The diagonal of (S_tile x Im_tile^T) gives
// per-row dot products; same trick with (S,S) and (Im,Im) gives the squared norms.
__global__ void cosdiag_wmma_kernel(const float* __restrict__ s,
                                    const float* __restrict__ im,
                                    float* __restrict__ partials,
                                    int numTiles) {
  extern __shared__ float smem[];
  const int lane = threadIdx.x & 31;
  const int wave = threadIdx.x >> 5;
  float* sL = smem + wave * (2 * TROWS * LSTRIDE);
  float* iL = sL + TROWS * LSTRIDE;

  const int wave0 = blockIdx.x * WAVES + wave;
  const int wstep = gridDim.x * WAVES;

  float acc = 0.f;

  for (int tile = wave0; tile < numTiles; tile += wstep) {  // wave-uniform branch: EXEC stays all-ones for WMMA
    const size_t rowBase = (size_t)tile * TROWS * DCOLS;
    v8f aSI = {}, aSS = {}, aII = {};

    // A-layout (== B-layout of the transpose): lanes 0-15 hold {K,K+1} of row=lane,
    // lanes 16-31 hold {K+2,K+3} of row=lane-16.
    const int koff = (lane >> 4) << 1;  // 0 or 2
    const int rsel = lane & 15;

    for (int c0 = 0; c0 < DCOLS; c0 += CHUNK) {
      // Stage 16 x CHUNK fp32 of each matrix: coalesced b128 global loads -> b128 LDS stores
#pragma unroll
      for (int it = 0; it < (TROWS * CHUNK / 4 / 32); ++it) {   // 8 iters
        const int f   = it * 32 + lane;
        const int row = f >> 4;          // CHUNK/4 = 16 float4 per row
        const int c4  = f & 15;
        const float4 sv = *(const float4*)(s  + rowBase + (size_t)row * DCOLS + c0 + c4 * 4);
        const float4 iv = *(const float4*)(im + rowBase + (size_t)row * DCOLS + c0 + c4 * 4);
        *(float4*)(sL + row * LSTRIDE + c4 * 4) = sv;
        *(float4*)(iL + row * LSTRIDE + c4 * 4) = iv;
      }
      asm volatile("s_wait_dscnt 0" ::: "memory");  // wave-local stage->compute handoff (no block barrier)

      const float* sRow = sL + rsel * LSTRIDE + koff;
      const float* iRow = iL + rsel * LSTRIDE + koff;
#pragma unroll
      for (int k = 0; k < CHUNK; k += 4) {
        v2f a = *(const v2f*)(sRow + k);   // ds_load_b64, bank-conflict-free (stride pad)
        v2f b = *(const v2f*)(iRow + k);
        aSI = __builtin_amdgcn_wmma_f32_16x16x4_f32(false, a, false, b, (short)0, aSI, false, false);
        aSS = __builtin_amdgcn_wmma_f32_16x16x4_f32(false, a, false, a, (short)0, aSS, false, false);
        aII = __builtin_amdgcn_wmma_f32_16x16x4_f32(false, b, false, b, (short)0, aII, false, false);
      }
      asm volatile("" ::: "memory");  // keep next chunk's LDS stores after these loads
    }

    // Diagonal lives at: M<8 -> (elem M, lane M); M>=8 -> (elem M-8, lane M+16)
    const bool ownerLo = lane < 8;
    const bool ownerHi = lane >= 24;
    const int  idx = ownerLo ? lane : ((lane - 24) & 7);
    const float dsi = pick8(aSI, idx);
    const float dss = pick8(aSS, idx);
    const float dii = pick8(aII, idx);
    const float EPS = 1.1e-13f;  // 1e-13 + 1e-14
    const float c = dsi / ((sqrtf(dss) + EPS) * (sqrtf(dii) + EPS));
    acc += (ownerLo || ownerHi) ? c : 0.f;
  }

  // wave reduction (wave32)
#pragma unroll
  for (int off = 16; off; off >>= 1) acc += __shfl_xor(acc, off, 32);

  __syncthreads();
  if (lane == 0) smem[wave] = acc;
  __syncthreads();
  if (threadIdx.x == 0) {
    float t = 0.f;
#pragma unroll
    for (int w = 0; w < WAVES; ++w) t += smem[w];
    partials[blockIdx.x] = t;
  }
}

// Deterministic fixed-order final reduction; writes loss = -sum(cos)
__global__ void final_reduce_kernel(const float* __restrict__ partials,
                                    float* __restrict__ out, int n) {
  __shared__ float sh[256];
  float t = 0.f;
  for (int i = threadIdx.x; i < n; i += 256) t += partials[i];
  sh[threadIdx.x] = t;
  __syncthreads();
#pragma unroll
  for (int s = 128; s; s >>= 1) {
    if ((int)threadIdx.x < s) sh[threadIdx.x] += sh[threadIdx.x + s];
    __syncthreads();
  }
  if (threadIdx.x == 0) out[0] = -sh[0];
}

extern "C" void kernel_launch(void* const* d_in, const int* in_sizes, int n_in,
                              void* d_out, int out_size, void* d_ws, size_t ws_size,
                              hipStream_t stream) {
  const float* s  = (const float*)d_in[0];
  const float* im = (const float*)d_in[1];
  // d_in[2] = temp (unused: diag(sim) loss is temperature-independent)
  float* out = (float*)d_out;
  float* partials = (float*)d_ws;

  const int N = in_sizes[0] / DCOLS;      // 65536
  const int numTiles = N / TROWS;         // 4096
  int grid = (numTiles + WAVES - 1) / WAVES;  // 1024 blocks -> one tile per wave
  if (grid < 1) grid = 1;
  if (grid > 4096) grid = 4096;           // grid-stride loop covers the rest

  const size_t smemBytes = (size_t)WAVES * 2 * TROWS * LSTRIDE * sizeof(float);  // 34,816 B

  cosdiag_wmma_kernel<<<grid, BLOCK, smemBytes, stream>>>(s, im, partials, numTiles);
  final_reduce_kernel<<<1, 256, 0, stream>>>(partials, out, grid);
}